// CoSSM_77206332113507
// MI455X (gfx1250) — compile-verified
//
#include <hip/hip_runtime.h>
#include <math.h>

// Problem constants from the reference (B=2, L=512).
#define BQ 2
#define LQ 512
#define BT (BQ * LQ)   // 1024 token rows, always the GEMM M dimension
#define DSTATE 16
#define DBL_LD 64      // padded row stride of the x_proj output buffer

typedef float v2f __attribute__((ext_vector_type(2)));
typedef float v8f __attribute__((ext_vector_type(8)));

__device__ __forceinline__ float siluf(float x) { return x / (1.f + __expf(-x)); }
__device__ __forceinline__ float softplusf(float x) { return (x > 20.f) ? x : log1pf(__expf(x)); }

// ---------------------------------------------------------------------------
// WMMA fp32 GEMM:  C[M,Npad] = act( A[M,K] * W[Nw,K]^T + bias )
// One wave per 16x16 C tile, V_WMMA_F32_16X16X4_F32 over K in steps of 4.
// Npad is a multiple of 16; for cols >= Nw the W row index is clamped (the
// garbage lands in pad columns of C that are never read). Fully branchless.
// When K % 32 == 0, the 16xK B-panel is staged through LDS with
// global_load_async_to_lds_b128 (CDNA5 async copy), shared by all 4 waves.
// act: 0 = none, 1 = silu, 2 = softplus
// ---------------------------------------------------------------------------
#define PANEL_LD 36    // floats; 144B row stride: 16B aligned, conflict-free-ish

__global__ void k_wmma_gemm(const float* __restrict__ A, int lda,
                            const float* __restrict__ W,
                            const float* __restrict__ bias,
                            float* __restrict__ C, int ldc,
                            int M, int Npad, int Nw, int K, int act)
{
    __shared__ float panel[16 * PANEL_LD];

    const int lane = threadIdx.x & 31;
    const int wave = threadIdx.x >> 5;
    const int n0 = blockIdx.x * 16;
    const int m0 = (blockIdx.y * 4 + wave) * 16;

    const int half = lane >> 4;        // 0: K lo pair, 1: K hi pair
    const int lm   = lane & 15;        // A-row / B-col within tile

    const float* arow = A + (size_t)(m0 + lm) * lda;

    v8f acc = {};

    if ((K & 31) == 0) {
        // ---- async-staged path: B panel chunks of 16x32 via LDS ----
        const int trow = threadIdx.x >> 3;   // 0..15 : W row within tile
        const int tseg = threadIdx.x & 7;    // 0..7  : 4-float segment
        const int wrS  = min(n0 + trow, Nw - 1);
        const float* gsrc = W + (size_t)wrS * K + tseg * 4;
        const unsigned ldsOff =
            (unsigned)(uintptr_t)&panel[trow * PANEL_LD + tseg * 4];
        const float* prow = &panel[lm * PANEL_LD];

        for (int k0 = 0; k0 < K; k0 += 32) {
            __syncthreads();   // previous chunk's reads complete
            asm volatile("global_load_async_to_lds_b128 %0, %1, off"
                         :: "v"(ldsOff), "v"(gsrc + k0) : "memory");
            asm volatile("s_wait_asynccnt 0x0" ::: "memory");
            __syncthreads();   // panel visible to all waves
            #pragma unroll
            for (int kk = 0; kk < 32; kk += 4) {
                const int kl = kk + half * 2;
                v2f a, b;
                a.x = arow[k0 + kl];
                a.y = arow[k0 + kl + 1];
                b.x = prow[kl];
                b.y = prow[kl + 1];
                acc = __builtin_amdgcn_wmma_f32_16x16x4_f32(
                    false, a, false, b, (short)0, acc, false, false);
            }
        }
    } else {
        // ---- direct path (small K: the dt projection, K = 8 or 16) ----
        const int wr = min(n0 + lm, Nw - 1);
        const float* brow = W + (size_t)wr * K;
        #pragma unroll 4
        for (int k = 0; k < K; k += 4) {
            const int kk = k + half * 2;
            v2f a, b;
            a.x = arow[kk];
            a.y = arow[kk + 1];
            b.x = brow[kk];
            b.y = brow[kk + 1];
            acc = __builtin_amdgcn_wmma_f32_16x16x4_f32(
                false, a, false, b, (short)0, acc, false, false);
        }
    }

    const int col = n0 + lm;
    const float bv = bias ? bias[col] : 0.f;
    #pragma unroll
    for (int r = 0; r < 8; ++r) {
        const int row = m0 + r + half * 8;  // ISA C/D layout: VGPR r, lane-half
        float v = acc[r] + bv;
        if (act == 1)      v = siluf(v);
        else if (act == 2) v = softplusf(v);
        C[(size_t)row * ldc + col] = v;
    }
    (void)M;
}

// ---------------------------------------------------------------------------
// conv1d k=3, pad (1,1): x[BT,Cin] (B,L,Cin layout) -> tmp[BT,Co]
// ---------------------------------------------------------------------------
__global__ void k_conv3(const float* __restrict__ x,
                        const float* __restrict__ w,   // [Co, Cin, 3]
                        float* __restrict__ tmp,
                        int Cin, int Co)
{
    const int i = blockIdx.x * blockDim.x + threadIdx.x;
    if (i >= BT * Co) return;
    const int o  = i % Co;
    const int bt = i / Co;
    const int b = bt / LQ, t = bt % LQ;
    const float* wr = w + (size_t)o * Cin * 3;
    float acc = 0.f;
    for (int c = 0; c < Cin; ++c) {
        const float* wc = wr + c * 3;
        #pragma unroll
        for (int k = 0; k < 3; ++k) {
            const int tt = t + k - 1;
            const float xv = (tt >= 0 && tt < LQ)
                ? x[((size_t)(b * LQ + tt)) * Cin + c] : 0.f;
            acc = fmaf(wc[k], xv, acc);
        }
    }
    tmp[i] = acc;
}

// Per-channel batch statistics over (B, L): block per channel.
__global__ void k_bn_stats(const float* __restrict__ tmp,
                           float* __restrict__ mean, float* __restrict__ rstd,
                           int Co)
{
    const int o = blockIdx.x;
    __shared__ float s1[256], s2[256];
    float a = 0.f, q = 0.f;
    for (int i = threadIdx.x; i < BT; i += blockDim.x) {
        const float v = tmp[(size_t)i * Co + o];
        a += v; q += v * v;
    }
    s1[threadIdx.x] = a; s2[threadIdx.x] = q;
    __syncthreads();
    for (int s = blockDim.x / 2; s > 0; s >>= 1) {
        if (threadIdx.x < s) { s1[threadIdx.x] += s1[threadIdx.x + s]; s2[threadIdx.x] += s2[threadIdx.x + s]; }
        __syncthreads();
    }
    if (threadIdx.x == 0) {
        const float m = s1[0] / (float)BT;
        const float var = s2[0] / (float)BT - m * m;
        mean[o] = m;
        rstd[o] = rsqrtf(var + 1e-5f);
    }
}

// normalize + relu + skip (identity or 1x1 conv) -> out[BT,Co]
__global__ void k_bn_relu_skip(const float* __restrict__ tmp,
                               const float* __restrict__ xin,
                               const float* __restrict__ mean,
                               const float* __restrict__ rstd,
                               const float* __restrict__ gamma,
                               const float* __restrict__ beta,
                               const float* __restrict__ skipw,  // [Co,Cin] or null
                               float* __restrict__ out,
                               int Cin, int Co)
{
    const int i = blockIdx.x * blockDim.x + threadIdx.x;
    if (i >= BT * Co) return;
    const int o  = i % Co;
    const int bt = i / Co;
    float v = (tmp[i] - mean[o]) * rstd[o] * gamma[o] + beta[o];
    v = fmaxf(v, 0.f);
    float sk;
    if (skipw) {
        const float* xr = xin + (size_t)bt * Cin;
        const float* wr = skipw + (size_t)o * Cin;
        float acc = 0.f;
        for (int c = 0; c < Cin; ++c) acc = fmaf(wr[c], xr[c], acc);
        sk = acc;
    } else {
        sk = xin[(size_t)bt * Cin + o];
    }
    out[i] = v + sk;
}

// depthwise causal conv k=4 (pad 3,0) + silu; reads xi view of xz[BT,2*di].
// rev=1: operate on time-flipped sequence (output kept in flipped order).
__global__ void k_dwconv_silu(const float* __restrict__ xz,
                              const float* __restrict__ cw,  // [di,1,4]
                              const float* __restrict__ cb,
                              float* __restrict__ xc,
                              int di, int rev)
{
    const int i = blockIdx.x * blockDim.x + threadIdx.x;
    if (i >= BT * di) return;
    const int d  = i % di;
    const int bt = i / di;
    const int b = bt / LQ, t = bt % LQ;
    const float* w4 = cw + (size_t)d * 4;
    float acc = cb[d];
    #pragma unroll
    for (int k = 0; k < 4; ++k) {
        const int tt = t + k - 3;
        if (tt < 0) continue;
        const int ts = rev ? (LQ - 1 - tt) : tt;
        acc = fmaf(w4[k], xz[((size_t)(b * LQ + ts)) * (2 * di) + d], acc);
    }
    xc[i] = siluf(acc);
}

// selective scan: thread per (b,d), 16-wide state in registers.
__global__ void k_scan(const float* __restrict__ u,     // [BT,di]
                       const float* __restrict__ dt,    // [BT,di]
                       const float* __restrict__ dbl,   // [BT,DBL_LD]; B at +dtr, C at +dtr+16
                       const float* __restrict__ A_log, // [di,16]
                       const float* __restrict__ Dv,    // [di]
                       float* __restrict__ y,
                       int di, int dtr)
{
    const int i = blockIdx.x * blockDim.x + threadIdx.x;
    if (i >= BQ * di) return;
    const int d = i % di;
    const int b = i / di;
    float An[DSTATE], h[DSTATE];
    #pragma unroll
    for (int n = 0; n < DSTATE; ++n) {
        An[n] = -__expf(A_log[(size_t)d * DSTATE + n]);
        h[n] = 0.f;
    }
    const float Dd = Dv[d];
    for (int t = 0; t < LQ; ++t) {
        const size_t r = (size_t)(b * LQ + t);
        const float dtv = dt[r * di + d];
        const float uv  = u[r * di + d];
        const float du  = dtv * uv;
        const float* row = dbl + r * DBL_LD + dtr;
        float yv = 0.f;
        #pragma unroll
        for (int n = 0; n < DSTATE; ++n) {
            h[n] = h[n] * __expf(dtv * An[n]) + du * row[n];
            yv = fmaf(h[n], row[DSTATE + n], yv);
        }
        y[r * di + d] = fmaf(uv, Dd, yv);
    }
}

// y = (y_f + flip(y_b)) * silu(z);  z is cols [di,2di) of xz.
__global__ void k_gate(const float* __restrict__ yf,
                       const float* __restrict__ yb,
                       const float* __restrict__ xz,
                       float* __restrict__ yc, int di)
{
    const int i = blockIdx.x * blockDim.x + threadIdx.x;
    if (i >= BT * di) return;
    const int d  = i % di;
    const int bt = i / di;
    const int b = bt / LQ, t = bt % LQ;
    const float zv = xz[(size_t)bt * (2 * di) + di + d];
    const float v = yf[i] + yb[((size_t)(b * LQ + (LQ - 1 - t))) * di + d];
    yc[i] = v * siluf(zv);
}

// out = xres + layernorm(m) * g + b   (block per token row)
__global__ void k_ln_res(const float* __restrict__ xres,
                         const float* __restrict__ m,
                         const float* __restrict__ g,
                         const float* __restrict__ bb,
                         float* __restrict__ out, int C)
{
    const int row = blockIdx.x;
    __shared__ float s1[256], s2[256];
    float a = 0.f, q = 0.f;
    for (int c = threadIdx.x; c < C; c += blockDim.x) {
        const float v = m[(size_t)row * C + c];
        a += v; q += v * v;
    }
    s1[threadIdx.x] = a; s2[threadIdx.x] = q;
    __syncthreads();
    for (int s = blockDim.x / 2; s > 0; s >>= 1) {
        if (threadIdx.x < s) { s1[threadIdx.x] += s1[threadIdx.x + s]; s2[threadIdx.x] += s2[threadIdx.x + s]; }
        __syncthreads();
    }
    const float mean = s1[0] / (float)C;
    const float var  = s2[0] / (float)C - mean * mean;
    const float rs   = rsqrtf(var + 1e-6f);
    for (int c = threadIdx.x; c < C; c += blockDim.x) {
        const size_t o = (size_t)row * C + c;
        out[o] = xres[o] + (m[o] - mean) * rs * g[c] + bb[c];
    }
}

// ---------------------------------------------------------------------------
// Host orchestration
// ---------------------------------------------------------------------------
struct MDir  { const float *conv_w, *conv_b, *W_x, *W_dt, *b_dt, *A_log, *D; };
struct MPar  { const float *W_in, *W_out; MDir fwd, bwd; };
struct LayerP {
    const float *e_conv, *g_conv, *e_bn_g, *e_bn_b, *g_bn_g, *g_bn_b;
    MPar me, mg;
    const float *ln1_g, *ln1_b, *ln2_g, *ln2_b;
    const float *e_skip, *g_skip;
};

extern "C" void kernel_launch(void* const* d_in, const int* in_sizes, int n_in,
                              void* d_out, int out_size, void* d_ws, size_t ws_size,
                              hipStream_t stream)
{
    (void)in_sizes; (void)n_in; (void)out_size; (void)ws_size;

    // ---- walk flattened params in setup_inputs() insertion order ----
    int idx = 0;
    auto nxtp = [&]() { return (const float*)d_in[idx++]; };
    const float* e_x = nxtp();
    const float* g_x = nxtp();

    const int ocs[3] = {128, 256, 512};
    const int ics[3] = {128, 128, 256};

    LayerP lp[3];
    for (int l = 0; l < 3; ++l) {
        LayerP& L = lp[l];
        L.e_conv = nxtp(); L.g_conv = nxtp();
        L.e_bn_g = nxtp(); L.e_bn_b = nxtp();
        L.g_bn_g = nxtp(); L.g_bn_b = nxtp();
        auto md = [&](MDir& dd) {
            dd.conv_w = nxtp(); dd.conv_b = nxtp(); dd.W_x = nxtp();
            dd.W_dt = nxtp();   dd.b_dt = nxtp();   dd.A_log = nxtp(); dd.D = nxtp();
        };
        auto mp = [&](MPar& m) { m.W_in = nxtp(); m.W_out = nxtp(); md(m.fwd); md(m.bwd); };
        mp(L.me); mp(L.mg);
        L.ln1_g = nxtp(); L.ln1_b = nxtp(); L.ln2_g = nxtp(); L.ln2_b = nxtp();
        L.e_skip = L.g_skip = nullptr;
        if (ics[l] != ocs[l]) { L.e_skip = nxtp(); L.g_skip = nxtp(); }
    }

    // ---- carve workspace (floats) ----
    float* w = (float*)d_ws;
    auto alloc = [&](size_t n) { float* p = w; w += n; return p; };
    const size_t AMAX = (size_t)BT * 512;
    float* actE[2] = { alloc(AMAX), alloc(AMAX) };
    float* actG[2] = { alloc(AMAX), alloc(AMAX) };
    float* tmp   = alloc(AMAX);
    float* meanb = alloc(512);
    float* rstdb = alloc(512);
    float* xz    = alloc((size_t)BT * 2048);
    float* xc_f  = alloc((size_t)BT * 1024);
    float* xc_b  = alloc((size_t)BT * 1024);
    float* dbl_f = alloc((size_t)BT * DBL_LD);
    float* dbl_b = alloc((size_t)BT * DBL_LD);
    float* dt_f  = alloc((size_t)BT * 1024);
    float* dt_b  = alloc((size_t)BT * 1024);
    float* y_f   = alloc((size_t)BT * 1024);
    float* y_b   = alloc((size_t)BT * 1024);
    float* y_c   = alloc((size_t)BT * 1024);
    float* m_out = alloc(AMAX);

    // C gets Npad = Nw rounded to 16 columns; caller guarantees ldc >= Npad.
    auto gemm = [&](const float* A, int lda, const float* W, const float* bias,
                    float* C, int ldc, int Nw, int K, int act) {
        const int Npad = (Nw + 15) & ~15;
        dim3 g((unsigned)(Npad / 16), (unsigned)(BT / 64));
        k_wmma_gemm<<<g, 128, 0, stream>>>(A, lda, W, bias, C, ldc,
                                           BT, Npad, Nw, K, act);
    };

    auto bimamba = [&](const float* x, const MPar& mp, int oc, float* mo) {
        const int di = 2 * oc;
        const int dtr = (oc + 15) / 16;   // 8 / 16 / 32
        const int ndbl = dtr + 2 * DSTATE;
        // in_proj: xz = x @ W_in^T   [BT, 2*di]
        gemm(x, oc, mp.W_in, nullptr, xz, 2 * di, 2 * di, oc, 0);
        const MDir* dirs[2] = { &mp.fwd, &mp.bwd };
        float* xcs[2]  = { xc_f,  xc_b  };
        float* dbls[2] = { dbl_f, dbl_b };
        float* dts[2]  = { dt_f,  dt_b  };
        float* ys[2]   = { y_f,   y_b   };
        for (int r = 0; r < 2; ++r) {
            const MDir& dd = *dirs[r];
            const int n1 = (BT * di + 255) / 256;
            k_dwconv_silu<<<n1, 256, 0, stream>>>(xz, dd.conv_w, dd.conv_b, xcs[r], di, r);
            // x_proj: dbl = xc @ W_x^T   [BT, ndbl] (padded row stride DBL_LD)
            gemm(xcs[r], di, dd.W_x, nullptr, dbls[r], DBL_LD, ndbl, di, 0);
            // dt = softplus(dt_r @ W_dt^T + b_dt)   [BT, di]
            gemm(dbls[r], DBL_LD, dd.W_dt, dd.b_dt, dts[r], di, di, dtr, 2);
            k_scan<<<(BQ * di + 255) / 256, 256, 0, stream>>>(
                xcs[r], dts[r], dbls[r], dd.A_log, dd.D, ys[r], di, dtr);
        }
        k_gate<<<(BT * di + 255) / 256, 256, 0, stream>>>(y_f, y_b, xz, y_c, di);
        // out_proj: mo = y @ W_out^T   [BT, oc]
        gemm(y_c, di, mp.W_out, nullptr, mo, oc, oc, di, 0);
    };

    const float* cur_e = e_x;
    const float* cur_g = g_x;
    int pp = 0;
    float* e_final = (float*)d_out;
    float* g_final = (float*)d_out + (size_t)BT * 512;

    for (int l = 0; l < 3; ++l) {
        const int ic = ics[l], oc = ocs[l];
        const int tot = BT * oc;
        const int nb = (tot + 255) / 256;
        const LayerP& L = lp[l];

        float* nxtE = actE[pp];
        float* nxtG = actG[pp];
        float* dstE = (l == 2) ? e_final : nxtE;
        float* dstG = (l == 2) ? g_final : nxtG;

        // ---- e stream ----
        k_conv3<<<nb, 256, 0, stream>>>(cur_e, L.e_conv, tmp, ic, oc);
        k_bn_stats<<<oc, 256, 0, stream>>>(tmp, meanb, rstdb, oc);
        k_bn_relu_skip<<<nb, 256, 0, stream>>>(tmp, cur_e, meanb, rstdb,
                                               L.e_bn_g, L.e_bn_b, L.e_skip, nxtE, ic, oc);
        bimamba(nxtE, L.me, oc, m_out);
        k_ln_res<<<BT, 256, 0, stream>>>(nxtE, m_out, L.ln1_g, L.ln1_b, dstE, oc);

        // ---- g stream ----
        k_conv3<<<nb, 256, 0, stream>>>(cur_g, L.g_conv, tmp, ic, oc);
        k_bn_stats<<<oc, 256, 0, stream>>>(tmp, meanb, rstdb, oc);
        k_bn_relu_skip<<<nb, 256, 0, stream>>>(tmp, cur_g, meanb, rstdb,
                                               L.g_bn_g, L.g_bn_b, L.g_skip, nxtG, ic, oc);
        bimamba(nxtG, L.mg, oc, m_out);
        k_ln_res<<<BT, 256, 0, stream>>>(nxtG, m_out, L.ln2_g, L.ln2_b, dstG, oc);

        cur_e = dstE;
        cur_g = dstG;
        pp ^= 1;
    }
}